// EnhancedHAN_82145544503903
// MI455X (gfx1250) — compile-verified
//
#include <hip/hip_runtime.h>
#include <math.h>

// Problem constants (match reference)
#define ND   40000
#define NS   30000
#define NE   300000
#define CC   256
#define HH   4
#define DDIM 64
#define OUTC 128

typedef float v2f __attribute__((ext_vector_type(2)));
typedef float v8f __attribute__((ext_vector_type(8)));

// ---------------- monotone float<->uint encoding for atomicMax on floats ----
__device__ __forceinline__ unsigned encf(float f) {
    unsigned u = __float_as_uint(f);
    return (u & 0x80000000u) ? ~u : (u | 0x80000000u);
}
__device__ __forceinline__ float decf(unsigned u) {
    return __uint_as_float((u & 0x80000000u) ? (u & 0x7FFFFFFFu) : ~u);
}
#define ENC_NEG_INF 0x007FFFFFu   // encf(-inf)

// ---------------- WMMA f32 GEMM: out[M,N] = op(A[M,K] @ B[K,N] + bias) ------
// Register-blocked: 256 threads = 8 waves arranged 2(M) x 4(N); each wave
// computes a 32x32 output block as 2x2 tiles of 16x16 -> 4 independent
// accumulator chains, A/B fragments each reused twice.
// Block tile: 64(M) x 128(N). A-row loads are clamped so EXEC stays all-ones
// through the WMMA loop; bounds are enforced only in the epilogue.
// mode 0: store (bias + optional addend)
// mode 1: relu store
// mode 2: tanh + atomic column-sum into colsum[N] (no store)
__global__ __launch_bounds__(256) void k_gemm_wmma(
    const float* __restrict__ A, const float* __restrict__ B,
    const float* __restrict__ bias, const float* __restrict__ addend,
    float* __restrict__ out, float* __restrict__ colsum,
    int M, int N, int K, int mode)
{
    const int lane  = threadIdx.x & 31;
    const int wave  = threadIdx.x >> 5;
    const int waveM = wave >> 2;        // 0..1
    const int waveN = wave & 3;         // 0..3
    const int half  = lane >> 4;        // 0: lanes 0-15, 1: lanes 16-31
    const int l16   = lane & 15;
    const int koff  = half * 2;

    const int mBase = blockIdx.y * 64 + waveM * 32;          // wave's first row
    const int nBase = blockIdx.x * 128 + waveN * 32;         // wave's first col

    // clamped A rows (keep loads in-bounds; garbage rows masked in epilogue)
    const int r0 = min(mBase + l16,      M - 1);
    const int r1 = min(mBase + 16 + l16, M - 1);
    const int c0 = nBase + l16;
    const int c1 = nBase + 16 + l16;

    v8f acc00 = {0.f,0.f,0.f,0.f,0.f,0.f,0.f,0.f};
    v8f acc01 = acc00, acc10 = acc00, acc11 = acc00;

    const float* __restrict__ A0 = A + (size_t)r0 * K;
    const float* __restrict__ A1 = A + (size_t)r1 * K;

    for (int k0 = 0; k0 < K; k0 += 4) {
        // A fragments: (k, k+1) contiguous -> single b64 load each
        v2f a0 = *(const v2f*)(A0 + k0 + koff);
        v2f a1 = *(const v2f*)(A1 + k0 + koff);
        // B fragments: two k-rows, stride N apart
        v2f b0, b1;
        b0.x = B[(size_t)(k0 + koff) * N + c0];
        b0.y = B[(size_t)(k0 + koff + 1) * N + c0];
        b1.x = B[(size_t)(k0 + koff) * N + c1];
        b1.y = B[(size_t)(k0 + koff + 1) * N + c1];
        acc00 = __builtin_amdgcn_wmma_f32_16x16x4_f32(false, a0, false, b0,
                                                      (short)0, acc00, false, false);
        acc01 = __builtin_amdgcn_wmma_f32_16x16x4_f32(false, a0, false, b1,
                                                      (short)0, acc01, false, false);
        acc10 = __builtin_amdgcn_wmma_f32_16x16x4_f32(false, a1, false, b0,
                                                      (short)0, acc10, false, false);
        acc11 = __builtin_amdgcn_wmma_f32_16x16x4_f32(false, a1, false, b1,
                                                      (short)0, acc11, false, false);
    }

    const v8f* accs[4] = {&acc00, &acc01, &acc10, &acc11};

    if (mode == 2) {
        const float bv0 = bias[c0];
        const float bv1 = bias[c1];
        float s0 = 0.f, s1 = 0.f;
        #pragma unroll
        for (int tm = 0; tm < 2; ++tm) {
            const int rbase = mBase + tm * 16 + half * 8;
            #pragma unroll
            for (int v = 0; v < 8; ++v) {
                if (rbase + v < M) {
                    s0 += tanhf((*accs[tm * 2 + 0])[v] + bv0);
                    s1 += tanhf((*accs[tm * 2 + 1])[v] + bv1);
                }
            }
        }
        atomicAdd(&colsum[c0], s0);
        atomicAdd(&colsum[c1], s1);
        return;
    }

    const float bv0 = bias ? bias[c0] : 0.f;
    const float bv1 = bias ? bias[c1] : 0.f;
    #pragma unroll
    for (int tm = 0; tm < 2; ++tm) {
        const int rbase = mBase + tm * 16 + half * 8;
        #pragma unroll
        for (int tn = 0; tn < 2; ++tn) {
            const v8f a = *accs[tm * 2 + tn];
            const int c = tn ? c1 : c0;
            const float bv = tn ? bv1 : bv0;
            #pragma unroll
            for (int v = 0; v < 8; ++v) {
                const int r = rbase + v;
                if (r < M) {
                    float val = a[v] + bv;
                    if (addend) val += addend[(size_t)r * N + c];
                    if (mode == 1) val = fmaxf(val, 0.f);
                    out[(size_t)r * N + c] = val;
                }
            }
        }
    }
}

// ---------------- small utility kernels -------------------------------------
__global__ void k_fill_u32(unsigned* __restrict__ p, unsigned v, int n) {
    int i = blockIdx.x * blockDim.x + threadIdx.x;
    if (i < n) p[i] = v;
}

// s[n,h] = sum_d h[n, h*64+d] * a[h*64+d]     (one thread per (node,head))
__global__ void k_node_scores(const float* __restrict__ h,
                              const float* __restrict__ a,
                              float* __restrict__ s, int n) {
    int t = blockIdx.x * blockDim.x + threadIdx.x;
    if (t >= n * HH) return;
    int node = t >> 2, hh = t & 3;
    const float* hp = h + (size_t)node * CC + hh * DDIM;
    const float* ap = a + hh * DDIM;
    float acc = 0.f;
    #pragma unroll 8
    for (int d = 0; d < DDIM; ++d) acc += hp[d] * ap[d];
    s[t] = acc;
}

// pass 1: alpha = leaky_relu(ssrc[src]+sdst[dst]); cache; atomicMax per (dst,h)
__global__ void k_edge_alpha_max(const float* __restrict__ ssrc,
                                 const float* __restrict__ sdst,
                                 const int* __restrict__ src,
                                 const int* __restrict__ dst,
                                 float* __restrict__ ebuf,
                                 unsigned* __restrict__ amax) {
    int t = blockIdx.x * blockDim.x + threadIdx.x;
    if (t >= NE * HH) return;
    int e = t >> 2, hh = t & 3;
    float a = ssrc[src[e] * HH + hh] + sdst[dst[e] * HH + hh];
    a = (a > 0.f) ? a : 0.2f * a;         // leaky_relu(0.2)
    ebuf[t] = a;
    atomicMax(&amax[dst[e] * HH + hh], encf(a));
}

// pass 2: e = exp(alpha - amax[dst]); cache; atomicAdd denom per (dst,h)
__global__ void k_edge_exp_denom(float* __restrict__ ebuf,
                                 const unsigned* __restrict__ amax,
                                 const int* __restrict__ dst,
                                 float* __restrict__ denom) {
    int t = blockIdx.x * blockDim.x + threadIdx.x;
    if (t >= NE * HH) return;
    int e = t >> 2, hh = t & 3;
    float m  = decf(amax[dst[e] * HH + hh]);
    float ev = expf(ebuf[t] - m);
    ebuf[t] = ev;
    atomicAdd(&denom[dst[e] * HH + hh], ev);
}

// pass 3: out[dst, c] += h_src[src, c] * e   (one block per edge, coalesced)
__global__ __launch_bounds__(256) void k_edge_scatter(
    const float* __restrict__ h_src, const float* __restrict__ ebuf,
    const int* __restrict__ src, const int* __restrict__ dst,
    float* __restrict__ outb) {
    int e = blockIdx.x;
    int c = threadIdx.x;
    int s = src[e], d = dst[e];
    float w = ebuf[e * HH + (c >> 6)];
    atomicAdd(&outb[(size_t)d * CC + c], h_src[(size_t)s * CC + c] * w);
}

// out = relu(acc / (denom + 1e-16))
__global__ void k_gat_finalize(float* __restrict__ outb,
                               const float* __restrict__ denom, int n_dst) {
    int t = blockIdx.x * blockDim.x + threadIdx.x;
    if (t >= n_dst * CC) return;
    int node = t >> 8, c = t & 255;
    float v = outb[t] / (denom[node * HH + (c >> 6)] + 1e-16f);
    outb[t] = fmaxf(v, 0.f);
}

// semantic attention: scores from column sums, 2-way softmax -> attn[2]
__global__ void k_group_score(const float* __restrict__ cs,
                              const float* __restrict__ q,
                              float invN, float* __restrict__ attn) {
    __shared__ float r0[256], r1[256];
    int c = threadIdx.x;
    r0[c] = cs[c] * q[c];
    r1[c] = cs[256 + c] * q[c];
    __syncthreads();
    for (int s = 128; s > 0; s >>= 1) {
        if (c < s) { r0[c] += r0[c + s]; r1[c] += r1[c + s]; }
        __syncthreads();
    }
    if (c == 0) {
        float s0 = r0[0] * invN, s1 = r1[0] * invN;
        float m = fmaxf(s0, s1);
        float e0 = expf(s0 - m), e1 = expf(s1 - m);
        float d = e0 + e1;
        attn[0] = e0 / d;
        attn[1] = e1 / d;
    }
}

// h = relu(attn0*o0 + attn1*o1), written in place into o0
__global__ void k_combine(float* __restrict__ o0, const float* __restrict__ o1,
                          const float* __restrict__ attn, int n) {
    int t = blockIdx.x * blockDim.x + threadIdx.x;
    if (t >= n) return;
    float v = attn[0] * o0[t] + attn[1] * o1[t];
    o0[t] = fmaxf(v, 0.f);
}

// ---------------- host-side helpers -----------------------------------------
static inline void launch_gemm(const float* A, const float* B, const float* bias,
                               const float* addend, float* out, float* colsum,
                               int M, int N, int K, int mode, hipStream_t s) {
    dim3 grid(N / 128, (M + 63) / 64), block(256);
    k_gemm_wmma<<<grid, block, 0, s>>>(A, B, bias, addend, out, colsum,
                                       M, N, K, mode);
}

static void run_gat(const float* h_src, const float* h_dst,
                    const float* a_s, const float* a_d,
                    const int* src, const int* dst,
                    int n_src, int n_dst, float* outb,
                    float* ssrc, float* sdst, unsigned* amax, float* denom,
                    float* ebuf, hipStream_t stream) {
    int t;
    t = n_src * HH;
    k_node_scores<<<(t + 255) / 256, 256, 0, stream>>>(h_src, a_s, ssrc, n_src);
    t = n_dst * HH;
    k_node_scores<<<(t + 255) / 256, 256, 0, stream>>>(h_dst, a_d, sdst, n_dst);
    k_fill_u32<<<(t + 255) / 256, 256, 0, stream>>>(amax, ENC_NEG_INF, t);
    hipMemsetAsync(denom, 0, (size_t)t * sizeof(float), stream);
    hipMemsetAsync(outb, 0, (size_t)n_dst * CC * sizeof(float), stream);

    int te = NE * HH;
    k_edge_alpha_max<<<(te + 255) / 256, 256, 0, stream>>>(ssrc, sdst, src, dst,
                                                           ebuf, amax);
    k_edge_exp_denom<<<(te + 255) / 256, 256, 0, stream>>>(ebuf, amax, dst, denom);
    k_edge_scatter<<<NE, 256, 0, stream>>>(h_src, ebuf, src, dst, outb);
    t = n_dst * CC;
    k_gat_finalize<<<(t + 255) / 256, 256, 0, stream>>>(outb, denom, n_dst);
}

extern "C" void kernel_launch(void* const* d_in, const int* in_sizes, int n_in,
                              void* d_out, int out_size, void* d_ws, size_t ws_size,
                              hipStream_t stream) {
    (void)in_sizes; (void)n_in; (void)out_size; (void)ws_size;

    const float* x_drug    = (const float*)d_in[0];
    const float* x_disease = (const float*)d_in[1];
    const int* ei_tr_s  = (const int*)d_in[2];
    const int* ei_tr_d  = (const int*)d_in[3];
    const int* ei_rv_s  = (const int*)d_in[4];
    const int* ei_rv_d  = (const int*)d_in[5];
    const int* ei_dd_s  = (const int*)d_in[6];
    const int* ei_dd_d  = (const int*)d_in[7];
    const int* ei_ss_s  = (const int*)d_in[8];
    const int* ei_ss_d  = (const int*)d_in[9];
    const float* Wp_drug = (const float*)d_in[10];
    const float* bp_drug = (const float*)d_in[11];
    const float* Wp_dis  = (const float*)d_in[12];
    const float* bp_dis  = (const float*)d_in[13];
    const float* a_s_tr  = (const float*)d_in[14];
    const float* a_d_tr  = (const float*)d_in[15];
    const float* a_s_rv  = (const float*)d_in[16];
    const float* a_d_rv  = (const float*)d_in[17];
    const float* a_s_dd  = (const float*)d_in[18];
    const float* a_d_dd  = (const float*)d_in[19];
    const float* a_s_ss  = (const float*)d_in[20];
    const float* a_d_ss  = (const float*)d_in[21];
    const float* Wk   = (const float*)d_in[22];
    const float* bk   = (const float*)d_in[23];
    const float* q    = (const float*)d_in[24];
    const float* Wd   = (const float*)d_in[25];
    const float* bd   = (const float*)d_in[26];
    const float* Wdis = (const float*)d_in[27];
    const float* bdis = (const float*)d_in[28];
    const float* Wr   = (const float*)d_in[29];
    const float* br   = (const float*)d_in[30];

    float* emb_out = (float*)d_out;           // [ND*128 | NS*128]

    // ---- workspace layout (floats) ----
    float* w = (float*)d_ws;
    size_t off = 0;
    float* hd    = w + off; off += (size_t)ND * CC;   // projected drug feats
    float* hs    = w + off; off += (size_t)NS * CC;   // projected disease feats
    float* o_tr  = w + off; off += (size_t)NS * CC;   // -> h_dis (in place)
    float* o_rev = w + off; off += (size_t)ND * CC;   // -> h_drug (in place)
    float* o_dd  = w + off; off += (size_t)ND * CC;
    float* o_ss  = w + off; off += (size_t)NS * CC;
    float* ebuf  = w + off; off += (size_t)NE * HH;   // per-edge alpha/e cache
    float* ssrc  = w + off; off += (size_t)ND * HH;
    float* sdst  = w + off; off += (size_t)ND * HH;
    unsigned* amax = (unsigned*)(w + off); off += (size_t)ND * HH;
    float* denom = w + off; off += (size_t)ND * HH;
    float* colsum = w + off; off += 512;
    float* attn   = w + off; off += 2;
    // residual buffers alias hd/hs (safe: only used after all GAT relations)
    float* res_d = hd;   // ND*128 <= ND*256
    float* res_s = hs;   // NS*128 <= NS*256

    // ---- 1) input projections (WMMA f32 GEMM) ----
    launch_gemm(x_drug,    Wp_drug, bp_drug, nullptr, hd, nullptr,
                ND, CC, CC, /*mode=*/0, stream);
    launch_gemm(x_disease, Wp_dis,  bp_dis,  nullptr, hs, nullptr,
                NS, CC, CC, /*mode=*/0, stream);

    // ---- 2) four GAT edge-softmax aggregations ----
    run_gat(hd, hs, a_s_tr, a_d_tr, ei_tr_s, ei_tr_d, ND, NS, o_tr,
            ssrc, sdst, amax, denom, ebuf, stream);
    run_gat(hs, hd, a_s_rv, a_d_rv, ei_rv_s, ei_rv_d, NS, ND, o_rev,
            ssrc, sdst, amax, denom, ebuf, stream);
    run_gat(hd, hd, a_s_dd, a_d_dd, ei_dd_s, ei_dd_d, ND, ND, o_dd,
            ssrc, sdst, amax, denom, ebuf, stream);
    run_gat(hs, hs, a_s_ss, a_d_ss, ei_ss_s, ei_ss_d, NS, NS, o_ss,
            ssrc, sdst, amax, denom, ebuf, stream);

    // ---- 3) semantic attention, drug group: [o_rev, o_dd] ----
    hipMemsetAsync(colsum, 0, 512 * sizeof(float), stream);
    launch_gemm(o_rev, Wk, bk, nullptr, nullptr, colsum,       ND, CC, CC, 2, stream);
    launch_gemm(o_dd,  Wk, bk, nullptr, nullptr, colsum + 256, ND, CC, CC, 2, stream);
    k_group_score<<<1, 256, 0, stream>>>(colsum, q, 1.f / (float)ND, attn);
    {
        int n = ND * CC;
        k_combine<<<(n + 255) / 256, 256, 0, stream>>>(o_rev, o_dd, attn, n);
    }   // h_drug now in o_rev

    // ---- 3b) semantic attention, disease group: [o_treats, o_ss] ----
    hipMemsetAsync(colsum, 0, 512 * sizeof(float), stream);
    launch_gemm(o_tr, Wk, bk, nullptr, nullptr, colsum,       NS, CC, CC, 2, stream);
    launch_gemm(o_ss, Wk, bk, nullptr, nullptr, colsum + 256, NS, CC, CC, 2, stream);
    k_group_score<<<1, 256, 0, stream>>>(colsum, q, 1.f / (float)NS, attn);
    {
        int n = NS * CC;
        k_combine<<<(n + 255) / 256, 256, 0, stream>>>(o_tr, o_ss, attn, n);
    }   // h_dis now in o_tr

    // ---- 4) residual projections: res = relu(x @ Wr + br) ----
    launch_gemm(x_drug,    Wr, br, nullptr, res_d, nullptr, ND, OUTC, CC, 1, stream);
    launch_gemm(x_disease, Wr, br, nullptr, res_s, nullptr, NS, OUTC, CC, 1, stream);

    // ---- 5) output heads: emb = h @ W + b + res ----
    launch_gemm(o_rev, Wd,   bd,   res_d, emb_out,                    nullptr,
                ND, OUTC, CC, 0, stream);
    launch_gemm(o_tr,  Wdis, bdis, res_s, emb_out + (size_t)ND * OUTC, nullptr,
                NS, OUTC, CC, 0, stream);
}